// EnhancedExternalMemory_78108275245208
// MI455X (gfx1250) — compile-verified
//
#include <hip/hip_runtime.h>
#include <hip/hip_bf16.h>
#include <stdint.h>

typedef float v2f  __attribute__((ext_vector_type(2)));
typedef float v8f  __attribute__((ext_vector_type(8)));
typedef unsigned int u32x4 __attribute__((ext_vector_type(4)));
typedef int   i32x4 __attribute__((ext_vector_type(4)));
typedef int   i32x8 __attribute__((ext_vector_type(8)));

#define NB 2048
#define NH 1024
#define NS 32768
#define ND 64

#define AST 1026                 // A LDS row stride (1024 + cl + pad)
#define BST 72                   // B tile LDS row stride (64 + 8 pad via TDM)
#define KCH 64                   // K rows per TDM tile
#define NCHUNK ((NS / 64) * (NH / KCH))   // 512 S-blocks * 16 K-chunks = 8192

// workspace float offsets
#define WS_WW    0               // [S]  ww_mean (atomic-accumulated)
#define WS_CMEAN 32768           // scalar c_mean
#define WS_CMN   32769           // [H] content mean
#define WS_PCM   33793           // [D] pc_mean
#define WS_TOTAL 33857

__device__ __forceinline__ unsigned ldsOffset(const void* p) {
  // generic pointer to LDS: low 32 bits are the LDS byte offset
  return (unsigned)(uintptr_t)p;
}

// TDM: DMA one [KCH x 64] fp32 tile of a [1025 x 32768] row-major tensor into
// LDS with an 8-dword pad after every 64 dwords (row stride 72 floats).
__device__ __forceinline__ void tdm_tile(unsigned lds_byte_off, const float* gptr) {
  unsigned long long ga = (unsigned long long)(uintptr_t)gptr;
  u32x4 g0;
  g0[0] = 1u;                                             // count=1 (user D#)
  g0[1] = lds_byte_off;                                   // lds_addr
  g0[2] = (unsigned)(ga & 0xffffffffull);                 // global_addr lo
  g0[3] = ((unsigned)(ga >> 32) & 0x1ffffffu) | (2u << 30); // addr hi | type=2
  i32x8 g1;
  g1[0] = (2 << 16) | (1 << 20) | (5 << 22) | (7 << 25);  // 4B elems, pad_en, 64dw interval, 8dw pad
  g1[1] = (int)0x80000000u;                               // tensor_dim0 = 32768 (lo16 << 16)
  g1[2] = (int)(0x0401u << 16);                           // tensor_dim1 = 1025 (lo16 << 16)
  g1[3] = (64 << 16);                                     // tile_dim0 = 64
  g1[4] = KCH;                                            // tile_dim1 = KCH, tile_dim2 = 0
  g1[5] = 32768;                                          // tensor_dim0_stride = S
  g1[6] = 0;
  g1[7] = 0;
  i32x4 gz; gz[0] = 0; gz[1] = 0; gz[2] = 0; gz[3] = 0;
  i32x8 gz8; gz8[0] = 0; gz8[1] = 0; gz8[2] = 0; gz8[3] = 0;
  gz8[4] = 0; gz8[5] = 0; gz8[6] = 0; gz8[7] = 0;
  __builtin_amdgcn_tensor_load_to_lds(g0, g1, gz, gz, gz8, 0);
}

__device__ __forceinline__ v8f wmma4(v2f a, v2f b, v8f c) {
  return __builtin_amdgcn_wmma_f32_16x16x4_f32(false, a, false, b, (short)0, c,
                                               false, false);
}

// Accumulate one K-chunk (KCH) of a 16x16 logits tile.
// A: LDS, row stride AST, already offset to this chunk's K base.
// Bt: LDS tile [KCH][BST]. row = lane&15, kh = (lane>>4)*2, col = wave*16+row.
__device__ __forceinline__ v8f qk_chunk(const float* A, const float* Bt,
                                        int row, int kh, int col, v8f acc) {
#pragma unroll
  for (int kk = 0; kk < KCH / 4; ++kk) {
    v2f a = *(const v2f*)(A + row * AST + kk * 4 + kh);
    v2f b;
    b.x = Bt[(kk * 4 + kh) * BST + col];
    b.y = Bt[(kk * 4 + kh + 1) * BST + col];
    acc = wmma4(a, b, acc);
  }
  return acc;
}

__device__ __forceinline__ void stageA(float* Alds, const float* q,
                                       const float* cl, int b0, int tid) {
  for (int i = tid; i < 16 * NH; i += 128) {
    int r = i >> 10, k = i & (NH - 1);
    Alds[r * AST + k] = q[(size_t)(b0 + r) * NH + k];
  }
  if (tid < 16) {
    Alds[tid * AST + 1024] = cl[b0 + tid];
    Alds[tid * AST + 1025] = 0.f;
  }
}

// ---------------------------------------------------------------------------
// Read path: flash softmax(QK) * strength, normalized, @ memory  -> [B, D]
// ---------------------------------------------------------------------------
__global__ __launch_bounds__(128) void read_flash_kernel(
    const float* __restrict__ query, const float* __restrict__ clevel,
    const float* __restrict__ memory, const float* __restrict__ strength,
    const float* __restrict__ W_read, const float* __restrict__ b_read,
    float* __restrict__ read_content) {
  __shared__ float Alds[16 * AST];
  __shared__ float Blds[2][KCH * BST];
  __shared__ float Pt[4][16 * 20];
  __shared__ float Ost[4][16 * 64];
  __shared__ float wsA[4][16];
  __shared__ float wsB[4][16];
  __shared__ float rowM[16];
  __shared__ float dInv[16];

  const int tid = threadIdx.x;
  const int lane = tid & 31;
  const int wave = tid >> 5;
  const int half = lane >> 4;
  const int kh = half * 2;
  const int row = lane & 15;
  const int wofs = wave * 16;
  const int b0 = blockIdx.x * 16;

  stageA(Alds, query, clevel, b0, tid);
  __syncthreads();

  float clr[8];
#pragma unroll
  for (int r = 0; r < 8; ++r) clr[r] = Alds[(half * 8 + r) * AST + 1024];

  // ---------------- pass 1: row maxima over S ----------------
  float m[8];
#pragma unroll
  for (int r = 0; r < 8; ++r) m[r] = -3.0e38f;

  if (wave == 0) {
    tdm_tile(ldsOffset(&Blds[0][0]), W_read);
    __builtin_amdgcn_s_wait_tensorcnt(0);
  }
  __syncthreads();

  v8f acc;
  for (int it = 0; it < NCHUNK; ++it) {
    const int kc = it & 15;
    const int sb = (it >> 4) * 64;
    if (wave == 0 && it + 1 < NCHUNK) {
      int kc2 = (it + 1) & 15, sb2 = ((it + 1) >> 4) * 64;
      tdm_tile(ldsOffset(&Blds[(it + 1) & 1][0]),
               W_read + (size_t)(kc2 * KCH) * NS + sb2);
    }
    if (kc == 0) {
#pragma unroll
      for (int r = 0; r < 8; ++r) acc[r] = 0.f;
    }
    acc = qk_chunk(Alds + kc * KCH, &Blds[it & 1][0], row, kh, wofs + row, acc);
    if (kc == 15) {
      const int s = sb + wofs + row;
      const float wtop = W_read[(size_t)1024 * NS + s];
      const float bias = b_read[s];
#pragma unroll
      for (int r = 0; r < 8; ++r) {
        float v = acc[r] + clr[r] * wtop + bias;
#pragma unroll
        for (int off = 8; off >= 1; off >>= 1)
          v = fmaxf(v, __shfl_xor(v, off, 32));
        m[r] = fmaxf(m[r], v);
      }
    }
    if (wave == 0) __builtin_amdgcn_s_wait_tensorcnt(0);
    __syncthreads();
  }

  if (row == 0) {
#pragma unroll
    for (int r = 0; r < 8; ++r) wsA[wave][half * 8 + r] = m[r];
  }
  __syncthreads();
  if (tid < 16)
    rowM[tid] = fmaxf(fmaxf(wsA[0][tid], wsA[1][tid]),
                      fmaxf(wsA[2][tid], wsA[3][tid]));
  __syncthreads();

  // ---------------- pass 2: P = exp(l-M)*str, sums, O += P @ memory --------
  float myM[8], se[8], ses[8];
#pragma unroll
  for (int r = 0; r < 8; ++r) {
    myM[r] = rowM[half * 8 + r];
    se[r] = 0.f;
    ses[r] = 0.f;
  }
  v8f O[4];
#pragma unroll
  for (int g = 0; g < 4; ++g)
#pragma unroll
    for (int r = 0; r < 8; ++r) O[g][r] = 0.f;

  if (wave == 0) {
    tdm_tile(ldsOffset(&Blds[0][0]), W_read);
    __builtin_amdgcn_s_wait_tensorcnt(0);
  }
  __syncthreads();

  for (int it = 0; it < NCHUNK; ++it) {
    const int kc = it & 15;
    const int sb = (it >> 4) * 64;
    if (wave == 0 && it + 1 < NCHUNK) {
      int kc2 = (it + 1) & 15, sb2 = ((it + 1) >> 4) * 64;
      tdm_tile(ldsOffset(&Blds[(it + 1) & 1][0]),
               W_read + (size_t)(kc2 * KCH) * NS + sb2);
    }
    if (kc == 0) {
#pragma unroll
      for (int r = 0; r < 8; ++r) acc[r] = 0.f;
    }
    acc = qk_chunk(Alds + kc * KCH, &Blds[it & 1][0], row, kh, wofs + row, acc);
    if (kc == 15) {
      const int s = sb + wofs + row;
      const float wtop = W_read[(size_t)1024 * NS + s];
      const float bias = b_read[s];
      const float str = strength[s];
#pragma unroll
      for (int r = 0; r < 8; ++r) {
        float p = __expf(acc[r] + clr[r] * wtop + bias - myM[r]);
        se[r] += p;
        float ps = p * str;
        ses[r] += ps;
        Pt[wave][(half * 8 + r) * 20 + row] = ps;  // per-wave buffer: no barrier
      }
      // O(16x64) += P(16x16) @ memory(16x64), K = this wave's 16 s-columns
#pragma unroll
      for (int kk = 0; kk < 4; ++kk) {
        v2f a = *(const v2f*)&Pt[wave][row * 20 + kk * 4 + kh];
        const float* mrow = memory + (size_t)(sb + wofs + kk * 4 + kh) * ND;
#pragma unroll
        for (int g = 0; g < 4; ++g) {
          v2f b;
          b.x = mrow[g * 16 + row];
          b.y = mrow[ND + g * 16 + row];
          O[g] = wmma4(a, b, O[g]);
        }
      }
    }
    if (wave == 0) __builtin_amdgcn_s_wait_tensorcnt(0);
    __syncthreads();
  }

#pragma unroll
  for (int r = 0; r < 8; ++r) {
#pragma unroll
    for (int off = 8; off >= 1; off >>= 1) {
      se[r] += __shfl_xor(se[r], off, 32);
      ses[r] += __shfl_xor(ses[r], off, 32);
    }
  }
  if (row == 0) {
#pragma unroll
    for (int r = 0; r < 8; ++r) {
      wsA[wave][half * 8 + r] = se[r];
      wsB[wave][half * 8 + r] = ses[r];
    }
  }
#pragma unroll
  for (int r = 0; r < 8; ++r) {
    int rr = half * 8 + r;
#pragma unroll
    for (int g = 0; g < 4; ++g) Ost[wave][rr * 64 + g * 16 + row] = O[g][r];
  }
  __syncthreads();
  if (tid < 16) {
    float seT = wsA[0][tid] + wsA[1][tid] + wsA[2][tid] + wsA[3][tid];
    float sesT = wsB[0][tid] + wsB[1][tid] + wsB[2][tid] + wsB[3][tid];
    dInv[tid] = 1.f / (sesT + 1e-6f * seT);
  }
  __syncthreads();
  for (int i = tid; i < 16 * 64; i += 128) {
    int r = i >> 6;
    float o = Ost[0][i] + Ost[1][i] + Ost[2][i] + Ost[3][i];
    read_content[(size_t)(b0 + r) * ND + (i & 63)] = o * dInv[r];
  }
}

// ---------------------------------------------------------------------------
// Write path: column-mean over B of softmax(QK_write) -> ww_mean[S] (atomics)
// ---------------------------------------------------------------------------
__global__ __launch_bounds__(128) void write_softmax_kernel(
    const float* __restrict__ query, const float* __restrict__ clevel,
    const float* __restrict__ W_write, const float* __restrict__ b_write,
    float* __restrict__ ww_sum) {
  __shared__ float Alds[16 * AST];
  __shared__ float Blds[2][KCH * BST];
  __shared__ float wsA[4][16];
  __shared__ float wsB[4][16];
  __shared__ float rowM[16];
  __shared__ float rowI[16];

  const int tid = threadIdx.x;
  const int lane = tid & 31;
  const int wave = tid >> 5;
  const int half = lane >> 4;
  const int kh = half * 2;
  const int row = lane & 15;
  const int wofs = wave * 16;
  const int b0 = blockIdx.x * 16;

  stageA(Alds, query, clevel, b0, tid);
  __syncthreads();

  float clr[8];
#pragma unroll
  for (int r = 0; r < 8; ++r) clr[r] = Alds[(half * 8 + r) * AST + 1024];

  // pass 1: online (max, sum-exp) per row
  float m[8], z[8];
#pragma unroll
  for (int r = 0; r < 8; ++r) {
    m[r] = -3.0e38f;
    z[r] = 0.f;
  }

  if (wave == 0) {
    tdm_tile(ldsOffset(&Blds[0][0]), W_write);
    __builtin_amdgcn_s_wait_tensorcnt(0);
  }
  __syncthreads();

  v8f acc;
  for (int it = 0; it < NCHUNK; ++it) {
    const int kc = it & 15;
    const int sb = (it >> 4) * 64;
    if (wave == 0 && it + 1 < NCHUNK) {
      int kc2 = (it + 1) & 15, sb2 = ((it + 1) >> 4) * 64;
      tdm_tile(ldsOffset(&Blds[(it + 1) & 1][0]),
               W_write + (size_t)(kc2 * KCH) * NS + sb2);
    }
    if (kc == 0) {
#pragma unroll
      for (int r = 0; r < 8; ++r) acc[r] = 0.f;
    }
    acc = qk_chunk(Alds + kc * KCH, &Blds[it & 1][0], row, kh, wofs + row, acc);
    if (kc == 15) {
      const int s = sb + wofs + row;
      const float wtop = W_write[(size_t)1024 * NS + s];
      const float bias = b_write[s];
#pragma unroll
      for (int r = 0; r < 8; ++r) {
        float v = acc[r] + clr[r] * wtop + bias;
        float tm = v;
#pragma unroll
        for (int off = 8; off >= 1; off >>= 1)
          tm = fmaxf(tm, __shfl_xor(tm, off, 32));
        float nm = fmaxf(m[r], tm);
        float e = __expf(v - nm);
#pragma unroll
        for (int off = 8; off >= 1; off >>= 1) e += __shfl_xor(e, off, 32);
        z[r] = z[r] * __expf(m[r] - nm) + e;
        m[r] = nm;
      }
    }
    if (wave == 0) __builtin_amdgcn_s_wait_tensorcnt(0);
    __syncthreads();
  }

  if (row == 0) {
#pragma unroll
    for (int r = 0; r < 8; ++r) {
      wsA[wave][half * 8 + r] = m[r];
      wsB[wave][half * 8 + r] = z[r];
    }
  }
  __syncthreads();
  if (tid < 16) {
    float M = fmaxf(fmaxf(wsA[0][tid], wsA[1][tid]),
                    fmaxf(wsA[2][tid], wsA[3][tid]));
    float Z = wsB[0][tid] * __expf(wsA[0][tid] - M) +
              wsB[1][tid] * __expf(wsA[1][tid] - M) +
              wsB[2][tid] * __expf(wsA[2][tid] - M) +
              wsB[3][tid] * __expf(wsA[3][tid] - M);
    rowM[tid] = M;
    rowI[tid] = 1.f / Z;
  }
  __syncthreads();

  float myM[8], myI[8];
#pragma unroll
  for (int r = 0; r < 8; ++r) {
    myM[r] = rowM[half * 8 + r];
    myI[r] = rowI[half * 8 + r];
  }

  // pass 2: accumulate column means with global fp32 atomics
  if (wave == 0) {
    tdm_tile(ldsOffset(&Blds[0][0]), W_write);
    __builtin_amdgcn_s_wait_tensorcnt(0);
  }
  __syncthreads();

  for (int it = 0; it < NCHUNK; ++it) {
    const int kc = it & 15;
    const int sb = (it >> 4) * 64;
    if (wave == 0 && it + 1 < NCHUNK) {
      int kc2 = (it + 1) & 15, sb2 = ((it + 1) >> 4) * 64;
      tdm_tile(ldsOffset(&Blds[(it + 1) & 1][0]),
               W_write + (size_t)(kc2 * KCH) * NS + sb2);
    }
    if (kc == 0) {
#pragma unroll
      for (int r = 0; r < 8; ++r) acc[r] = 0.f;
    }
    acc = qk_chunk(Alds + kc * KCH, &Blds[it & 1][0], row, kh, wofs + row, acc);
    if (kc == 15) {
      const int s = sb + wofs + row;
      const float wtop = W_write[(size_t)1024 * NS + s];
      const float bias = b_write[s];
      float colsum = 0.f;
#pragma unroll
      for (int r = 0; r < 8; ++r)
        colsum += __expf(acc[r] + clr[r] * wtop + bias - myM[r]) * myI[r];
      colsum += __shfl_xor(colsum, 16, 32);
      if (lane < 16) unsafeAtomicAdd(&ww_sum[s], colsum * (1.f / (float)NB));
    }
    if (wave == 0) __builtin_amdgcn_s_wait_tensorcnt(0);
    __syncthreads();
  }
}

// ---------------------------------------------------------------------------
// consciousness_boost = read_content @ W_ce + b_ce   (K = 64, WMMA)
// ---------------------------------------------------------------------------
__global__ __launch_bounds__(128) void boost_kernel(
    const float* __restrict__ rc, const float* __restrict__ W_ce,
    const float* __restrict__ b_ce, float* __restrict__ out_boost) {
  __shared__ float Ards[16 * 68];
  const int tid = threadIdx.x;
  const int lane = tid & 31;
  const int wave = tid >> 5;
  const int half = lane >> 4;
  const int kh = half * 2;
  const int row = lane & 15;
  const int b0 = blockIdx.x * 16;
  const int h0 = blockIdx.y * 64;
  for (int i = tid; i < 16 * 64; i += 128) {
    int r = i >> 6, k = i & 63;
    Ards[r * 68 + k] = rc[(size_t)(b0 + r) * ND + k];
  }
  __syncthreads();
  const int col = h0 + wave * 16 + row;
  v8f acc;
#pragma unroll
  for (int r = 0; r < 8; ++r) acc[r] = 0.f;
#pragma unroll
  for (int kk = 0; kk < 16; ++kk) {
    v2f a = *(const v2f*)&Ards[row * 68 + kk * 4 + kh];
    v2f b;
    b.x = W_ce[(size_t)(kk * 4 + kh) * NH + col];
    b.y = W_ce[(size_t)(kk * 4 + kh + 1) * NH + col];
    acc = wmma4(a, b, acc);
  }
  const float bias = b_ce[col];
#pragma unroll
  for (int r = 0; r < 8; ++r)
    out_boost[(size_t)(b0 + half * 8 + r) * NH + col] = acc[r] + bias;
}

// ---------------------------------------------------------------------------
// small kernels
// ---------------------------------------------------------------------------
__global__ void ws_init_kernel(float* __restrict__ ws) {
  int i = blockIdx.x * 256 + threadIdx.x;
  if (i < WS_TOTAL) ws[i] = 0.f;
}

__global__ void mean_kernel(const float* __restrict__ content,
                            const float* __restrict__ clevel,
                            float* __restrict__ ws) {
  int h = blockIdx.x * 128 + threadIdx.x;
  float s = 0.f;
  for (int b = 0; b < NB; ++b) s += content[(size_t)b * NH + h];
  ws[WS_CMN + h] = s * (1.f / (float)NB);
  if (blockIdx.x == 0) {
    __shared__ float red[128];
    float cs = 0.f;
    for (int b = threadIdx.x; b < NB; b += 128) cs += clevel[b];
    red[threadIdx.x] = cs;
    __syncthreads();
    for (int st = 64; st >= 1; st >>= 1) {
      if (threadIdx.x < st) red[threadIdx.x] += red[threadIdx.x + st];
      __syncthreads();
    }
    if (threadIdx.x == 0) ws[WS_CMEAN] = red[0] * (1.f / (float)NB);
  }
}

__global__ void pcmean_kernel(const float* __restrict__ W_cp,
                              const float* __restrict__ b_cp,
                              float* __restrict__ ws) {
  int d = threadIdx.x;
  const float* cm = ws + WS_CMN;
  float s = b_cp[d];
  for (int h = 0; h < NH; ++h) s += cm[h] * W_cp[h * ND + d];
  ws[WS_PCM + d] = s;
}

__global__ void memory_update_kernel(
    const float* __restrict__ memory, const float* __restrict__ age,
    const float* __restrict__ strength, const float* __restrict__ ws,
    float* __restrict__ out_mem, float* __restrict__ out_age,
    float* __restrict__ out_str) {
  int i = blockIdx.x * 256 + threadIdx.x;
  if (i >= NS * ND) return;
  int s = i >> 6, d = i & 63;
  float c_mean = ws[WS_CMEAN];
  float w = ws[WS_WW + s] * (c_mean * 2.f);
  bool mask = (w > 0.005f) && (c_mean >= 0.3f);
  float cons = 1.f / (1.f + __expf(-(age[s] * 0.1f + c_mean)));
  float alpha = w * cons;
  float mv = memory[i];
  out_mem[i] = mask ? (1.f - alpha) * mv + alpha * ws[WS_PCM + d] : mv;
  if (d == 0) {
    out_age[s] = age[s] + (mask ? 1.f : 0.f);
    out_str[s] =
        mask ? fminf(fmaxf(strength[s] + w * 0.1f, 0.1f), 2.f) : strength[s];
  }
}

// ---------------------------------------------------------------------------
extern "C" void kernel_launch(void* const* d_in, const int* in_sizes, int n_in,
                              void* d_out, int out_size, void* d_ws,
                              size_t ws_size, hipStream_t stream) {
  (void)in_sizes; (void)n_in; (void)out_size; (void)ws_size;
  const float* query   = (const float*)d_in[0];
  const float* clevel  = (const float*)d_in[1];
  const float* content = (const float*)d_in[2];
  const float* memory  = (const float*)d_in[3];
  const float* mem_age = (const float*)d_in[4];
  const float* mem_str = (const float*)d_in[5];
  const float* W_read  = (const float*)d_in[6];
  const float* b_read  = (const float*)d_in[7];
  const float* W_write = (const float*)d_in[8];
  const float* b_write = (const float*)d_in[9];
  const float* W_cp    = (const float*)d_in[10];
  const float* b_cp    = (const float*)d_in[11];
  const float* W_ce    = (const float*)d_in[12];
  const float* b_ce    = (const float*)d_in[13];

  float* out = (float*)d_out;
  float* ws  = (float*)d_ws;

  float* out_rc    = out;                              // [B, D]
  float* out_boost = out + (size_t)NB * ND;            // [B, H]
  float* out_mem   = out_boost + (size_t)NB * NH;      // [S, D]
  float* out_age   = out_mem + (size_t)NS * ND;        // [S]
  float* out_str   = out_age + NS;                     // [S]

  hipLaunchKernelGGL(ws_init_kernel, dim3((WS_TOTAL + 255) / 256), dim3(256),
                     0, stream, ws);
  hipLaunchKernelGGL(mean_kernel, dim3(NH / 128), dim3(128), 0, stream,
                     content, clevel, ws);
  hipLaunchKernelGGL(pcmean_kernel, dim3(1), dim3(64), 0, stream, W_cp, b_cp,
                     ws);
  hipLaunchKernelGGL(read_flash_kernel, dim3(NB / 16), dim3(128), 0, stream,
                     query, clevel, memory, mem_str, W_read, b_read, out_rc);
  hipLaunchKernelGGL(boost_kernel, dim3(NB / 16, NH / 64), dim3(128), 0,
                     stream, out_rc, W_ce, b_ce, out_boost);
  hipLaunchKernelGGL(write_softmax_kernel, dim3(NB / 16), dim3(128), 0, stream,
                     query, clevel, W_write, b_write, ws + WS_WW);
  hipLaunchKernelGGL(memory_update_kernel, dim3((NS * ND + 255) / 256),
                     dim3(256), 0, stream, memory, mem_age, mem_str, ws,
                     out_mem, out_age, out_str);
}